// PoolingConnection_61237643707061
// MI455X (gfx1250) — compile-verified
//
#include <hip/hip_runtime.h>
#include <math.h>

// Overlapping max-pool: window 64x64, stride 6, clip-at-edge (== -inf pad).
// In:  s [128, 384, 384] fp32   Out: [128, 64, 64] fp32
// Decomposition: 64-window @stride6 = 10 full 6-blocks + first-4 of block+10,
// applied separably in x then y. One workgroup per feature map; each wave
// streams 48 rows via async global->LDS b128 with double buffering.

#define H_DIM 384
#define W_DIM 384
#define POOL_S 6
#define OW 64            // W_DIM / POOL_S
#define OH 64            // H_DIM / POOL_S
#define FULLB 10         // full 6-blocks per 64-window
#define NWAVE 8
#define ROWS_PER_WAVE (H_DIM / NWAVE)   // 48 (multiple of POOL_S)

#ifndef __has_builtin
#define __has_builtin(x) 0
#endif

#if defined(__gfx1250__) && __has_builtin(__builtin_amdgcn_global_load_async_to_lds_b128)
#define HAVE_ASYNC_LDS 1
#else
#define HAVE_ASYNC_LDS 0
#if defined(__gfx1250__)
#warning "gfx1250 async-to-LDS builtin unavailable; using sync float4 fallback"
#endif
#endif

#if defined(__gfx1250__) && __has_builtin(__builtin_amdgcn_s_wait_asynccnt)
#define WAIT_ASYNC(n) __builtin_amdgcn_s_wait_asynccnt(n)
#else
#define WAIT_ASYNC(n) asm volatile("s_wait_asynccnt %0" ::"i"(n) : "memory")
#endif

struct __align__(16) Smem {
  float stage[NWAVE][2][W_DIM];   // 24 KB: per-wave double-buffered row staging
  float xsc[NWAVE][2 * OW];       //  4 KB: [0..63]=block max, [64..127]=first-4 partial
  float ybm[OH][OW];              // 16 KB: y-block (6-row) maxima
  float ypt[OH][OW];              // 16 KB: y-block first-4-row maxima
};                                // total 61440 B

#if HAVE_ASYNC_LDS
// Builtin signature (clang-22, gfx1250): (v4i addrspace(1)*, v4i addrspace(3)*, Imm, Imm)
typedef int v4i __attribute__((vector_size(16)));
typedef __attribute__((address_space(1))) v4i gv4i;
typedef __attribute__((address_space(3))) v4i lv4i;
__device__ __forceinline__ void row_async_issue(const float* g, float* l) {
  // 3 x b128 per lane covers 384 floats (lane*16B within each 512B chunk,
  // imm offset is added to BOTH global and LDS addresses).
  __builtin_amdgcn_global_load_async_to_lds_b128((gv4i*)g, (lv4i*)l, 0, 0);
  __builtin_amdgcn_global_load_async_to_lds_b128((gv4i*)g, (lv4i*)l, 512, 0);
  __builtin_amdgcn_global_load_async_to_lds_b128((gv4i*)g, (lv4i*)l, 1024, 0);
}
#endif

__global__ __launch_bounds__(256) void pool_fused(const float* __restrict__ s,
                                                  float* __restrict__ out) {
  __shared__ Smem sm;
  const int k = blockIdx.x;
  const int tid = (int)threadIdx.x;
  const int wave = tid >> 5;   // 0..7 (wave32)
  const int lane = tid & 31;
  const float* gK = s + (size_t)k * (H_DIM * W_DIM);

  const int yStart = wave * ROWS_PER_WAVE;
  float bacc0 = -INFINITY, bacc1 = -INFINITY;   // 6-row block max (j=lane, j=lane+32)
  float pacc0 = -INFINITY, pacc1 = -INFINITY;   // first-4-row partial max

#if HAVE_ASYNC_LDS
  row_async_issue(gK + (size_t)yStart * W_DIM + lane * 4,
                  &sm.stage[wave][0][lane * 4]);
#endif

  for (int r = 0; r < ROWS_PER_WAVE; ++r) {
    const int y = yStart + r;
#if HAVE_ASYNC_LDS
    float* cur = sm.stage[wave][r & 1];
    if (r + 1 < ROWS_PER_WAVE) {
      row_async_issue(gK + (size_t)(y + 1) * W_DIM + lane * 4,
                      &sm.stage[wave][(r + 1) & 1][lane * 4]);
      WAIT_ASYNC(3);   // current row's 3 loads done; next row's 3 stay in flight
    } else {
      WAIT_ASYNC(0);
    }
    asm volatile("" ::: "memory");
#else
    float* cur = sm.stage[wave][0];
    const float4* g4 = (const float4*)(gK + (size_t)y * W_DIM);
    float4* l4 = (float4*)cur;
    l4[lane] = g4[lane];
    l4[lane + 32] = g4[lane + 32];
    l4[lane + 64] = g4[lane + 64];
    asm volatile("" ::: "memory");
#endif

    // x-block maxima: lane owns blocks b=lane and b=lane+32.
    // LDS banks: (6*lane) mod 64 = 2*(3*lane mod 32) -> 32 distinct banks.
    {
      const float* r0 = cur + 6 * lane;
      const float* r1 = cur + 6 * (lane + 32);
      float p0 = fmaxf(fmaxf(r0[0], r0[1]), fmaxf(r0[2], r0[3]));
      float m0 = fmaxf(p0, fmaxf(r0[4], r0[5]));
      float p1 = fmaxf(fmaxf(r1[0], r1[1]), fmaxf(r1[2], r1[3]));
      float m1 = fmaxf(p1, fmaxf(r1[4], r1[5]));
      sm.xsc[wave][lane] = m0;
      sm.xsc[wave][OW + lane] = p0;
      sm.xsc[wave][lane + 32] = m1;
      sm.xsc[wave][OW + lane + 32] = p1;
    }
    // Same-wave DS ops are in-order (DScnt); only stop compiler reordering.
    asm volatile("" ::: "memory");

    // Row sliding max. j0 = lane (<=31): j0+10 <= 41, never clipped.
    // j1 = lane+32: clip blocks >= 64 (window runs past x=383).
    float rm0, rm1;
    {
      const float* xb = sm.xsc[wave];
      const int j0 = lane, j1 = lane + 32;
      float m = xb[j0];
#pragma unroll
      for (int d = 1; d < FULLB; ++d) m = fmaxf(m, xb[j0 + d]);
      m = fmaxf(m, xb[OW + j0 + FULLB]);
      rm0 = m;
      float n = xb[j1];
#pragma unroll
      for (int d = 1; d < FULLB; ++d) {
        float v = xb[j1 + d];                 // index <= 72 < 128: in xsc row
        n = (j1 + d < OW) ? fmaxf(n, v) : n;  // clipped -> discard
      }
      if (j1 + FULLB < OW) n = fmaxf(n, xb[OW + j1 + FULLB]);
      rm1 = n;
    }

    const int yin = r % POOL_S;               // uniform per wave
    bacc0 = fmaxf(bacc0, rm0);
    bacc1 = fmaxf(bacc1, rm1);
    if (yin < 4) { pacc0 = fmaxf(pacc0, rm0); pacc1 = fmaxf(pacc1, rm1); }
    if (yin == POOL_S - 1) {
      const int yb = y / POOL_S;              // wave*8 .. wave*8+7
      sm.ybm[yb][lane] = bacc0;
      sm.ybm[yb][lane + 32] = bacc1;
      sm.ypt[yb][lane] = pacc0;
      sm.ypt[yb][lane + 32] = pacc1;
      bacc0 = bacc1 = pacc0 = pacc1 = -INFINITY;
    }
  }

  __syncthreads();

  // y sliding max over block maxima; coalesced 128B stores per wave.
  const int j = tid & (OW - 1);
  const int i0 = tid >> 6;                    // uniform per wave
#pragma unroll
  for (int t = 0; t < 16; ++t) {
    const int i = i0 + 4 * t;
    float m = sm.ybm[i][j];
#pragma unroll
    for (int d = 1; d < FULLB; ++d)
      if (i + d < OH) m = fmaxf(m, sm.ybm[i + d][j]);
    if (i + FULLB < OH) m = fmaxf(m, sm.ypt[i + FULLB][j]);
    out[(((size_t)k * OH) + i) * OW + j] = m;
  }
}

extern "C" void kernel_launch(void* const* d_in, const int* in_sizes, int n_in,
                              void* d_out, int out_size, void* d_ws, size_t ws_size,
                              hipStream_t stream) {
  (void)n_in; (void)out_size; (void)d_ws; (void)ws_size;
  const float* s = (const float*)d_in[0];
  float* out = (float*)d_out;
  const int F = in_sizes[0] / (H_DIM * W_DIM);  // 128
  pool_fused<<<dim3(F), dim3(256), 0, stream>>>(s, out);
}